// MemorySpatialAttention_68908455297100
// MI455X (gfx1250) — compile-verified
//
#include <hip/hip_runtime.h>

// MI455X / gfx1250, wave32. Feature GEMM runs on V_WMMA_F32_16X16X4_F32
// (fp32-exact; workload is bandwidth-bound at ~40MB total traffic vs 23.3TB/s,
// so low-precision WMMA buys nothing -- f32 matrix path is the right choice).

typedef __attribute__((ext_vector_type(2))) float v2f;
typedef __attribute__((ext_vector_type(4))) float v4f;
typedef __attribute__((ext_vector_type(8))) float v8f;

#define B_   8
#define N_   2048
#define L_   56      // C*L, C==1
#define H_   128
#define NB   17
#define HALFW 8      // NB/2
#define ROWS (B_ * N_)   // 16384

// ---------------------------------------------------------------------------
// Kernel 1: features = LeakyReLU(BN(x @ conv_w^T + b))  for both inputs.
//   x tile:   [16384, 56] ; conv_w: [128, 56] (row-major) ; out: [16384, 128]
//   grid = (256, 2): 256 row-blocks of 64 rows; y selects input vs memory.
//   block = 128 threads = 4 waves; each wave -> 16 rows x 128 cols,
//   14 k-steps (K=56) x 8 col tiles of v_wmma_f32_16x16x4_f32.
// ---------------------------------------------------------------------------
__global__ __launch_bounds__(128) void msa_features_kernel(
    const float* __restrict__ x_in, const float* __restrict__ x_mem,
    const float* __restrict__ conv_w, const float* __restrict__ conv_b,
    const float* __restrict__ bn_gamma, const float* __restrict__ bn_beta,
    const float* __restrict__ bn_mean, const float* __restrict__ bn_var,
    float* __restrict__ feats)          // [2, 16384, 128]
{
    __shared__ float w_lds[H_ * L_];    // 28 KB of 320 KB/WGP

    const int tid = threadIdx.x;
    for (int i = tid; i < H_ * L_; i += 128) w_lds[i] = conv_w[i];
    __syncthreads();

    const int which = blockIdx.y;                       // 0=input, 1=state_memory
    const float* __restrict__ src = which ? x_mem : x_in;
    float* __restrict__ dst = feats + (size_t)which * ((size_t)ROWS * H_);

    const int wave   = tid >> 5;
    const int lane   = tid & 31;
    const int lane16 = lane & 15;
    const int half   = lane >> 4;                        // 0|1: K/M high half
    const int row_base = blockIdx.x * 64 + wave * 16;    // [0, 16384)

    v8f acc[8];
    #pragma unroll
    for (int t = 0; t < 8; ++t) acc[t] = v8f{};

    // A fragment (16x4 f32): lane16 -> M row; VGPR0/1 hold K = 2*half, 2*half+1
    const float* __restrict__ arow = src + (size_t)(row_base + lane16) * L_;

    for (int kk = 0; kk < 14; ++kk) {                    // K = 56 = 14 * 4
        const int k = kk * 4 + 2 * half;
        v2f a;
        a.x = arow[k];
        a.y = arow[k + 1];
        #pragma unroll
        for (int t = 0; t < 8; ++t) {                    // 8 col tiles of H=128
            const int col = t * 16 + lane16;             // B[k][n] = conv_w[n][k]
            v2f b;
            b.x = w_lds[col * L_ + k];
            b.y = w_lds[col * L_ + k + 1];
            acc[t] = __builtin_amdgcn_wmma_f32_16x16x4_f32(
                false, a, false, b, (short)0, acc[t], false, false);
        }
    }

    // Fused epilogue: conv bias + eval-BN + LeakyReLU(0.1).
    // D layout: VGPR r, lanes 0-15 -> M=r, lanes 16-31 -> M=r+8; N = lane16.
    #pragma unroll
    for (int t = 0; t < 8; ++t) {
        const int col   = t * 16 + lane16;
        const float scl = bn_gamma[col] * rsqrtf(bn_var[col] + 1e-5f);
        const float sft = bn_beta[col] - bn_mean[col] * scl;
        const float bia = conv_b[col];
        #pragma unroll
        for (int r = 0; r < 8; ++r) {
            const int row = row_base + r + 8 * half;
            float h = (acc[t][r] + bia) * scl + sft;
            h = (h >= 0.0f) ? h : 0.1f * h;
            dst[(size_t)row * H_ + col] = h;
        }
    }
}

// ---------------------------------------------------------------------------
// Kernel 2: banded softmax attention + residual blend. One wave32 per row.
//   sim[i,j] over the 17 clamped neighbors (duplicates at the edges dedup,
//   matching mask[i, clamp(i+w)] = 1 assignment semantics), softmax, then
//   out = 0.5*x + 0.5*(attn @ mem_flat).
// ---------------------------------------------------------------------------
__global__ __launch_bounds__(256) void msa_attn_kernel(
    const float* __restrict__ x_in, const float* __restrict__ x_mem,
    const float* __restrict__ feats, float* __restrict__ out)
{
    const int wave = threadIdx.x >> 5;
    const int lane = threadIdx.x & 31;
    const int gw   = blockIdx.x * 8 + wave;     // row in [0, 16384)
    const int b    = gw / N_;
    const int n    = gw - b * N_;

    const float* __restrict__ f_in  = feats;
    const float* __restrict__ f_mem = feats + (size_t)ROWS * H_;

    // lane holds 4 of the 128 feature channels
    const v4f fi = *(const v4f*)(f_in + (size_t)gw * H_ + lane * 4);

    float sims[NB];
    int   cols[NB];
    int   prev = -1;
    #pragma unroll
    for (int w = 0; w < NB; ++w) {
        int j = n - HALFW + w;
        j = (j < 0) ? 0 : ((j > N_ - 1) ? N_ - 1 : j);
        cols[w] = j;
        if (j != prev) {                         // clamp duplicates are contiguous
            const v4f fm = *(const v4f*)(f_mem + ((size_t)b * N_ + j) * H_ + lane * 4);
            float p = fi.x * fm.x + fi.y * fm.y + fi.z * fm.z + fi.w * fm.w;
            #pragma unroll
            for (int off = 16; off > 0; off >>= 1)   // wave32 reduction
                p += __shfl_xor(p, off, 32);
            sims[w] = p;
        } else {
            sims[w] = -__builtin_inff();         // dedup'd duplicate -> excluded
        }
        prev = j;
    }

    float m = sims[0];
    #pragma unroll
    for (int w = 1; w < NB; ++w) m = fmaxf(m, sims[w]);

    float wt[NB];
    float denom = 0.0f;
    #pragma unroll
    for (int w = 0; w < NB; ++w) {
        const float e = (sims[w] == -__builtin_inff()) ? 0.0f : __expf(sims[w] - m);
        wt[w] = e;
        denom += e;
    }
    const float inv = 1.0f / denom;

    // out[l] = 0.5*x[l] + 0.5 * sum_w wt[w]/denom * mem[cols[w]][l],  l in [0,56)
    for (int l = lane; l < L_; l += 32) {
        float acc = 0.0f;
        #pragma unroll
        for (int w = 0; w < NB; ++w)
            acc += wt[w] * x_mem[((size_t)b * N_ + cols[w]) * L_ + l];
        const size_t o = (size_t)gw * L_ + l;
        out[o] = 0.5f * x_in[o] + 0.5f * (acc * inv);
    }
}

// ---------------------------------------------------------------------------
extern "C" void kernel_launch(void* const* d_in, const int* in_sizes, int n_in,
                              void* d_out, int out_size, void* d_ws, size_t ws_size,
                              hipStream_t stream) {
    (void)in_sizes; (void)n_in; (void)out_size; (void)ws_size;
    const float* x_in   = (const float*)d_in[0];
    const float* x_mem  = (const float*)d_in[1];
    const float* conv_w = (const float*)d_in[2];
    const float* conv_b = (const float*)d_in[3];
    const float* gamma  = (const float*)d_in[4];
    const float* beta   = (const float*)d_in[5];
    const float* mean   = (const float*)d_in[6];
    const float* var    = (const float*)d_in[7];

    float* feats = (float*)d_ws;   // needs 2 * 16384 * 128 * 4 = 16 MB scratch

    dim3 g1(ROWS / 64, 2);         // 256 row-blocks x {input, memory}
    msa_features_kernel<<<g1, 128, 0, stream>>>(
        x_in, x_mem, conv_w, conv_b, gamma, beta, mean, var, feats);

    msa_attn_kernel<<<ROWS / 8, 256, 0, stream>>>(
        x_in, x_mem, feats, (float*)d_out);
}